// TemporalClassifier_26663156973887
// MI455X (gfx1250) — compile-verified
//
#include <hip/hip_runtime.h>
#include <hip/hip_bf16.h>

#define N_NODES 100000
#define N_EDGES 1600000
#define T_STEPS 64
#define D_H 256
#define D_OUT 16

typedef __attribute__((ext_vector_type(2))) float v2f;
typedef __attribute__((ext_vector_type(8))) float v8f;

// ---------------------------------------------------------------------------
// Degree kernels: deg = segment_sum(ones over dst) + 1 ; dinv = rsqrt(deg)
// ---------------------------------------------------------------------------
__global__ __launch_bounds__(256) void deg_init_kernel(float* __restrict__ deg, int n) {
    int i = blockIdx.x * 256 + threadIdx.x;
    if (i < n) deg[i] = 1.0f;  // self-loop
}

__global__ __launch_bounds__(256) void deg_count_kernel(const int* __restrict__ dst,
                                                        float* __restrict__ deg, int e) {
    int i = blockIdx.x * 256 + threadIdx.x;
    if (i < e) {
        int d = __builtin_nontemporal_load(&dst[i]);  // streamed once: NT
        __hip_atomic_fetch_add(&deg[d], 1.0f, __ATOMIC_RELAXED,
                               __HIP_MEMORY_SCOPE_AGENT);
    }
}

__global__ __launch_bounds__(256) void deg_rsqrt_kernel(float* __restrict__ deg, int n) {
    int i = blockIdx.x * 256 + threadIdx.x;
    if (i < n) deg[i] = rsqrtf(deg[i]);
}

// ---------------------------------------------------------------------------
// Dense GEMM: C[M,256] = act(A)[M,256] @ B[256,256], V_WMMA_F32_16X16X4_F32.
// One block = 4 waves = one 16-row strip x full 256 cols.
// A strip (16x256 f32, 16KB) staged to LDS via global_load_async_to_lds_b128
// (ASYNCcnt) then read as ds_load_b64 fragments; each A fragment feeds 4
// WMMAs (wave owns 16x64 via 4 accumulators). RELU_A applies ReLU to A reads
// (used for layer 2 so the standalone ReLU pass over 100MB disappears).
//
// Frag layout (ISA 7.12.2, f32 16x16x4):
//   A: lane L -> row m0+(L&15), K = k + ((L>>4)<<1) .. +1   (v2f)
//   B: lane L -> col n+(L&15),  same K split                 (v2f)
//   D: elem r -> row m0 + r + ((L>>4)<<3), col n+(L&15)      (v8f)
// ---------------------------------------------------------------------------
template <bool RELU_A>
__global__ __launch_bounds__(128) void gemm_wmma_f32_kernel(const float* __restrict__ A,
                                                            const float* __restrict__ B,
                                                            float* __restrict__ C) {
    const int K = 256, NC = 256;
    __shared__ float sA[16 * 256];  // 16 KB

    const int tid = threadIdx.x;
    const int wave = tid >> 5;
    const int lane = tid & 31;
    const int m0 = blockIdx.x * 16;
    const int l15 = lane & 15;
    const int khalf = (lane >> 4) << 1;  // 0 or 2

    // --- async stage of A strip into LDS (8 x b128 per lane) ---
    {
        const char* gsrc = (const char*)(A + (size_t)m0 * K);
        const unsigned ldsBase = (unsigned)(size_t)(&sA[0]);
#pragma unroll
        for (int it = 0; it < 8; ++it) {
            const int idx = it * 128 + tid;  // float4 index within strip
            unsigned ldsAddr = ldsBase + (unsigned)idx * 16u;
            const char* g = gsrc + (size_t)idx * 16;
            asm volatile("global_load_async_to_lds_b128 %0, %1, off"
                         :: "v"(ldsAddr), "v"(g)
                         : "memory");
        }
        asm volatile("s_wait_asynccnt 0x0" ::: "memory");
    }
    __syncthreads();

    const int n0 = wave * 64;  // 4 tiles: n0, n0+16, n0+32, n0+48
    v8f acc0 = {}, acc1 = {}, acc2 = {}, acc3 = {};

    const float* aL = &sA[l15 * 256 + khalf];
    const float* bp = B + (size_t)khalf * NC + (n0 + l15);

#pragma unroll 2
    for (int k = 0; k < K; k += 4) {
        v2f a;
        a.x = aL[0];
        a.y = aL[1];
        if (RELU_A) {
            a.x = fmaxf(a.x, 0.0f);
            a.y = fmaxf(a.y, 0.0f);
        }
        v2f b;
        b.x = bp[0];
        b.y = bp[NC];
        acc0 = __builtin_amdgcn_wmma_f32_16x16x4_f32(false, a, false, b, (short)0, acc0, false, false);
        b.x = bp[16];
        b.y = bp[NC + 16];
        acc1 = __builtin_amdgcn_wmma_f32_16x16x4_f32(false, a, false, b, (short)0, acc1, false, false);
        b.x = bp[32];
        b.y = bp[NC + 32];
        acc2 = __builtin_amdgcn_wmma_f32_16x16x4_f32(false, a, false, b, (short)0, acc2, false, false);
        b.x = bp[48];
        b.y = bp[NC + 48];
        acc3 = __builtin_amdgcn_wmma_f32_16x16x4_f32(false, a, false, b, (short)0, acc3, false, false);
        aL += 4;
        bp += 4 * NC;
    }

    const int rbase = m0 + ((lane >> 4) << 3);
    float* cptr = C + (size_t)rbase * NC + (n0 + l15);
#pragma unroll
    for (int r = 0; r < 8; ++r) {
        cptr[(size_t)r * NC + 0]  = acc0[r];
        cptr[(size_t)r * NC + 16] = acc1[r];
        cptr[(size_t)r * NC + 32] = acc2[r];
        cptr[(size_t)r * NC + 48] = acc3[r];
    }
}

// ---------------------------------------------------------------------------
// GCN init: out[i,h] = xw[i,h] * dinv[i]^2 + bias[h]   (self-loop + bias term)
// ---------------------------------------------------------------------------
__global__ __launch_bounds__(256) void gcn_init_kernel(const float* __restrict__ xw,
                                                       const float* __restrict__ dinv,
                                                       const float* __restrict__ bias,
                                                       float* __restrict__ out) {
    const int i = blockIdx.x;
    const int h = threadIdx.x;
    const float di = dinv[i];
    const size_t idx = (size_t)i * D_H + h;
    out[idx] = xw[idx] * di * di + bias[h];
}

// ---------------------------------------------------------------------------
// Edge scatter: out[dst] += xw[src] * dinv[src]*dinv[dst]
// 4 edges/block; 64 threads/edge, float4 per thread; no-return f32 atomics.
// Edge-index streams use NT loads to keep the L2-resident xw buffer hot.
// ---------------------------------------------------------------------------
__global__ __launch_bounds__(256) void edge_scatter_kernel(const float* __restrict__ xw,
                                                           const int* __restrict__ src,
                                                           const int* __restrict__ dst,
                                                           const float* __restrict__ dinv,
                                                           float* __restrict__ out, int e) {
    const int eidx = blockIdx.x * 4 + (threadIdx.x >> 6);
    if (eidx >= e) return;
    const int h4 = (threadIdx.x & 63) << 2;
    const int s = __builtin_nontemporal_load(&src[eidx]);
    const int d = __builtin_nontemporal_load(&dst[eidx]);
    const float coef = dinv[s] * dinv[d];
    const float4 v = *(const float4*)(xw + (size_t)s * D_H + h4);
    float* o = out + (size_t)d * D_H + h4;
    __hip_atomic_fetch_add(o + 0, v.x * coef, __ATOMIC_RELAXED, __HIP_MEMORY_SCOPE_AGENT);
    __hip_atomic_fetch_add(o + 1, v.y * coef, __ATOMIC_RELAXED, __HIP_MEMORY_SCOPE_AGENT);
    __hip_atomic_fetch_add(o + 2, v.z * coef, __ATOMIC_RELAXED, __HIP_MEMORY_SCOPE_AGENT);
    __hip_atomic_fetch_add(o + 3, v.w * coef, __ATOMIC_RELAXED, __HIP_MEMORY_SCOPE_AGENT);
}

// ---------------------------------------------------------------------------
// Segment mean-pool with fused ReLU: pooled[t,h] = mean(relu(h)) over
// nodes in [ptr[t], ptr[t+1]). Grid = T blocks, thread = hidden channel.
// ---------------------------------------------------------------------------
__global__ __launch_bounds__(256) void pool_relu_kernel(const float* __restrict__ h,
                                                        const int* __restrict__ ptr,
                                                        float* __restrict__ pooled) {
    const int t = blockIdx.x;
    const int j = threadIdx.x;
    const int lo = ptr[t];
    const int hi = ptr[t + 1];
    float s = 0.0f;
    for (int i = lo; i < hi; ++i) s += fmaxf(h[(size_t)i * D_H + j], 0.0f);
    const float cnt = (float)(hi - lo);
    pooled[(size_t)t * D_H + j] = s / fmaxf(cnt, 1.0f);
}

// ---------------------------------------------------------------------------
// GRU over T steps + final linear head. One block of 256 threads; h in LDS.
// Gate order (r, z, n); gx = x @ w_ih^T + b_ih, gh = h @ w_hh^T + b_hh.
// ---------------------------------------------------------------------------
__global__ __launch_bounds__(256) void gru_head_kernel(const float* __restrict__ pooled,
                                                       const float* __restrict__ w_ih,
                                                       const float* __restrict__ w_hh,
                                                       const float* __restrict__ b_ih,
                                                       const float* __restrict__ b_hh,
                                                       const float* __restrict__ Wc,
                                                       const float* __restrict__ bc,
                                                       float* __restrict__ out) {
    __shared__ float sx[D_H];
    __shared__ float sh[D_H];
    const int j = threadIdx.x;
    sh[j] = 0.0f;
    __syncthreads();

    for (int t = 0; t < T_STEPS; ++t) {
        sx[j] = pooled[(size_t)t * D_H + j];
        __syncthreads();

        float gx[3], gh[3];
#pragma unroll
        for (int g = 0; g < 3; ++g) {
            const int gi = g * D_H + j;
            const float* wi = w_ih + (size_t)gi * D_H;
            const float* wh = w_hh + (size_t)gi * D_H;
            float ax = b_ih[gi];
            float ah = b_hh[gi];
            for (int k = 0; k < D_H; ++k) {
                ax = fmaf(sx[k], wi[k], ax);
                ah = fmaf(sh[k], wh[k], ah);
            }
            gx[g] = ax;
            gh[g] = ah;
        }
        const float r = 1.0f / (1.0f + expf(-(gx[0] + gh[0])));
        const float z = 1.0f / (1.0f + expf(-(gx[1] + gh[1])));
        const float n = tanhf(gx[2] + r * gh[2]);
        const float hnew = (1.0f - z) * n + z * sh[j];
        __syncthreads();
        sh[j] = hnew;
        __syncthreads();
    }

    if (j < D_OUT) {
        const float* wc = Wc + (size_t)j * D_H;
        float acc = bc[j];
        for (int k = 0; k < D_H; ++k) acc = fmaf(sh[k], wc[k], acc);
        out[j] = acc;
    }
}

// ---------------------------------------------------------------------------
// Launch
// ---------------------------------------------------------------------------
extern "C" void kernel_launch(void* const* d_in, const int* in_sizes, int n_in,
                              void* d_out, int out_size, void* d_ws, size_t ws_size,
                              hipStream_t stream) {
    const float* x    = (const float*)d_in[0];
    const int*   edge = (const int*)d_in[1];
    const int*   ptr  = (const int*)d_in[2];
    const float* W1   = (const float*)d_in[3];
    const float* b1   = (const float*)d_in[4];
    const float* W2   = (const float*)d_in[5];
    const float* b2   = (const float*)d_in[6];
    const float* w_ih = (const float*)d_in[7];
    const float* w_hh = (const float*)d_in[8];
    const float* b_ih = (const float*)d_in[9];
    const float* b_hh = (const float*)d_in[10];
    const float* Wc   = (const float*)d_in[11];
    const float* bc   = (const float*)d_in[12];
    float* out = (float*)d_out;

    const int* srcI = edge;
    const int* dstI = edge + N_EDGES;

    // Workspace layout (256B aligned slices)
    char* ws = (char*)d_ws;
    size_t off = 0;
    auto take = [&](size_t bytes) -> char* {
        char* p = ws + off;
        off = (off + bytes + 255) & ~(size_t)255;
        return p;
    };
    float* dinv   = (float*)take((size_t)N_NODES * sizeof(float));
    float* bufA   = (float*)take((size_t)N_NODES * D_H * sizeof(float));  // xw
    float* bufB   = (float*)take((size_t)N_NODES * D_H * sizeof(float));  // h (pre-ReLU)
    float* pooled = (float*)take((size_t)T_STEPS * D_H * sizeof(float));
    (void)ws_size;

    const int nBlk = (N_NODES + 255) / 256;
    const int eBlk = (N_EDGES + 255) / 256;

    // Degrees
    deg_init_kernel<<<nBlk, 256, 0, stream>>>(dinv, N_NODES);
    deg_count_kernel<<<eBlk, 256, 0, stream>>>(dstI, dinv, N_EDGES);
    deg_rsqrt_kernel<<<nBlk, 256, 0, stream>>>(dinv, N_NODES);

    const int gemmGrid = N_NODES / 16;  // 6250 strips

    // ---- GCN layer 1 ----  xw1 = x @ W1
    gemm_wmma_f32_kernel<false><<<gemmGrid, 128, 0, stream>>>(x, W1, bufA);
    gcn_init_kernel<<<N_NODES, 256, 0, stream>>>(bufA, dinv, b1, bufB);
    edge_scatter_kernel<<<N_EDGES / 4, 256, 0, stream>>>(bufA, srcI, dstI, dinv, bufB, N_EDGES);

    // ---- GCN layer 2 ----  xw2 = relu(h1) @ W2  (ReLU fused into A reads)
    gemm_wmma_f32_kernel<true><<<gemmGrid, 128, 0, stream>>>(bufB, W2, bufA);
    gcn_init_kernel<<<N_NODES, 256, 0, stream>>>(bufA, dinv, b2, bufB);
    edge_scatter_kernel<<<N_EDGES / 4, 256, 0, stream>>>(bufA, srcI, dstI, dinv, bufB, N_EDGES);

    // ---- Pool (fused ReLU) + GRU + head ----
    pool_relu_kernel<<<T_STEPS, 256, 0, stream>>>(bufB, ptr, pooled);
    gru_head_kernel<<<1, 256, 0, stream>>>(pooled, w_ih, w_hh, b_ih, b_hh, Wc, bc, out);

    (void)in_sizes; (void)n_in; (void)out_size;
}